// Jitter_2370821947465
// MI455X (gfx1250) — compile-verified
//
#include <hip/hip_runtime.h>

// Jitter along time axis: out[b,c,t] = q[b,c,idx[t]],
// idx[t] = (replace_rand[t] < 12) ? clamped(t +/- 1) : t
//
// B=32, C=512, T=4096  ->  16384 rows of 4096 f32 (16 KiB each).
// Pure bandwidth problem: 512 MiB HBM traffic, floor ~22us at 23.3 TB/s.
// Strategy: one block per row; async-DMA the whole row into LDS
// (GLOBAL_LOAD_ASYNC_TO_LDS_B128, ASYNCcnt), resolve the +/-1 gather in LDS,
// store 128-bit aligned float4 results.

#define TT 4096
#define PROB 12

typedef __attribute__((ext_vector_type(4))) float f4;
typedef __attribute__((ext_vector_type(4))) int   i4;

// gcc-style vector to match the builtin's expected pointee type
// ('__attribute__((__vector_size__(4 * sizeof(int)))) int').
typedef int v4i __attribute__((vector_size(16)));

// Detect gfx1250 async-copy builtins (device pass only; host pass lacks them
// and takes the plain fallback, which is fine since host never codegens this).
#if defined(__has_builtin)
#if __has_builtin(__builtin_amdgcn_global_load_async_to_lds_b128)
#define USE_ASYNC_LDS 1
#endif
#endif
#ifndef USE_ASYNC_LDS
#define USE_ASYNC_LDS 0
#endif

#if USE_ASYNC_LDS
typedef __attribute__((address_space(1))) v4i* as1_v4i;   // global (__device__)
typedef __attribute__((address_space(3))) v4i* as3_v4i;   // LDS (__shared__)
#endif

__global__ __launch_bounds__(256) void jitter_kernel(
    const float* __restrict__ q,
    const int* __restrict__ rep,
    const int* __restrict__ dir,
    float* __restrict__ out)
{
    __shared__ float s_row[TT];                     // 16 KiB: one full row
    const int tid = threadIdx.x;
    const long long base = (long long)blockIdx.x * TT;
    const float* __restrict__ qrow = q + base;

    // ---- Stage the row into LDS (128-bit granules, fully coalesced) ----
#if USE_ASYNC_LDS
#pragma unroll
    for (int k = 0; k < 4; ++k) {
        const int v4 = k * 256 + tid;               // which float4 of the row
        __builtin_amdgcn_global_load_async_to_lds_b128(
            (as1_v4i)(qrow + v4 * 4),
            (as3_v4i)(&s_row[v4 * 4]),
            /*offset=*/0, /*cpol=*/0);
    }
#if __has_builtin(__builtin_amdgcn_s_wait_asynccnt)
    __builtin_amdgcn_s_wait_asynccnt(0);
#else
    asm volatile("s_wait_asynccnt 0" ::: "memory");
#endif
#else
    // Fallback: vector load -> LDS store (still b128 both sides).
#pragma unroll
    for (int k = 0; k < 4; ++k) {
        const int v4 = k * 256 + tid;
        *(f4*)(&s_row[v4 * 4]) = *(const f4*)(qrow + v4 * 4);
    }
#endif
    __syncthreads();

    // ---- Gather in LDS, write aligned float4 results ----
#pragma unroll
    for (int k = 0; k < 4; ++k) {
        const int v4 = k * 256 + tid;
        const int t0 = v4 * 4;
        const i4 r = *(const i4*)(rep + t0);        // L2-resident (16 KiB hot set)
        const i4 d = *(const i4*)(dir + t0);
        f4 o;
#pragma unroll
        for (int j = 0; j < 4; ++j) {
            const int t  = t0 + j;
            int nb = t + ((d[j] == 0) ? -1 : 1);
            if (t == 0)      nb = 1;
            if (t == TT - 1) nb = TT - 2;
            const int idx = (r[j] < PROB) ? nb : t;
            o[j] = s_row[idx];                      // ds_load_b32, +/-1 only
        }
        *(f4*)(out + base + t0) = o;                // global_store_b128
    }
}

extern "C" void kernel_launch(void* const* d_in, const int* in_sizes, int n_in,
                              void* d_out, int out_size, void* d_ws, size_t ws_size,
                              hipStream_t stream) {
    const float* q  = (const float*)d_in[0];   // quantized [B,C,T] f32
    const int* rep  = (const int*)d_in[1];     // replace_rand [T] i32
    const int* dir  = (const int*)d_in[2];     // dir_rand     [T] i32
    float* out      = (float*)d_out;           // [B,C,T] f32

    const int rows = in_sizes[0] / TT;         // B*C = 16384
    jitter_kernel<<<rows, 256, 0, stream>>>(q, rep, dir, out);
}